// CLIP_3DAdapter_22840636080315
// MI455X (gfx1250) — compile-verified
//
#include <hip/hip_runtime.h>
#include <hip/hip_bf16.h>
#include <math.h>

#if defined(__HIP_DEVICE_COMPILE__)
#if !__has_builtin(__builtin_amdgcn_wmma_f32_16x16x4_f32)
#error "gfx1250 f32 WMMA builtin __builtin_amdgcn_wmma_f32_16x16x4_f32 not available"
#endif
#endif

typedef __attribute__((ext_vector_type(2))) float v2f;
typedef __attribute__((ext_vector_type(8))) float v8f;

#define GRID 48
#define GG3  (GRID * GRID * GRID)   // 110592
#define VV   8
#define HH   27
#define WW   27
#define PP   (VV * HH * WW)         // 5832 points
#define CC   1152
#define CD   288

// ---------------------------------------------------------------------------
// WMMA f32 GEMM:  D[M,N] = A[M,K] * B[K,N]  (+ Cin if non-null)
//
// Each wave computes a 16 x (16*NT) output strip:
//   - NT independent v8f accumulators -> NT independent v_wmma chains per
//     K-step (RAW distance >= NT covers the WMMA->WMMA hazard, ISA 7.12.1).
//   - The A fragment is loaded once per K-step and reused by all NT WMMAs.
//   - All NT B fragments are materialized as distinct values before the WMMA
//     burst so the scheduler can batch the loads and issue WMMAs back-to-back
//     instead of serializing each WMMA on its own load.
//
// Fragment layouts (CDNA5 ISA 7.12.2, wave32):
//   A 16x4 : lanes 0-15 -> M=lane, lanes 16-31 -> M=lane-16;
//            VGPR j holds K = 2*(lane>>4) + j
//   B 4x16 : lanes hold N = lane&15; VGPR j holds K = 2*(lane>>4) + j
//   C/D    : VGPR i: M = i + 8*(lane>>4), N = lane&15
// ---------------------------------------------------------------------------
template <int NT>
__global__ void gemm_wmma_f32(const float* __restrict__ A,
                              const float* __restrict__ B,
                              const float* __restrict__ Cin,
                              float* __restrict__ D,
                              int M, int N, int K)
{
    const int lane = threadIdx.x & 31;
    const int wave = threadIdx.x >> 5;
    const int wavesPerBlock = blockDim.x >> 5;
    const int tile = blockIdx.x * wavesPerBlock + wave;

    const int nGroups = N / (16 * NT);     // N assumed multiple of 16*NT
    const int mTiles  = (M + 15) >> 4;
    if (tile >= mTiles * nGroups) return;  // wave-uniform exit

    const int mBase = (tile / nGroups) << 4;
    const int nBase = (tile % nGroups) * (16 * NT);

    const int hi = lane >> 4;    // 0 | 1
    const int lo = lane & 15;

    const int  aRow  = mBase + lo;
    const bool aOk   = aRow < M;
    const int  aRowS = aOk ? aRow : 0;     // clamp row, zero the fragment

    const float* __restrict__ aBase = A + (size_t)aRowS * K + 2 * hi;
    const float* __restrict__ bBase = B + (size_t)(2 * hi) * N + nBase + lo;

    v8f acc[NT];
#pragma unroll
    for (int t = 0; t < NT; ++t) acc[t] = (v8f){};

    for (int k0 = 0; k0 < K; k0 += 4) {
        // A fragment: two contiguous floats per lane, shared by all NT tiles
        float a0 = aBase[k0];
        float a1 = aBase[k0 + 1];

        // Batch-load all NT B fragments into distinct registers first
        const float* bp = bBase + (size_t)k0 * N;
        float bx[NT], by[NT];
#pragma unroll
        for (int t = 0; t < NT; ++t) {
            bx[t] = bp[t * 16];
            by[t] = bp[t * 16 + N];
        }

        a0 = aOk ? a0 : 0.0f;
        a1 = aOk ? a1 : 0.0f;
        v2f a; a.x = a0; a.y = a1;

#pragma unroll
        for (int t = 0; t < NT; ++t) {
            v2f b; b.x = bx[t]; b.y = by[t];
            acc[t] = __builtin_amdgcn_wmma_f32_16x16x4_f32(
                /*neg_a=*/false, a, /*neg_b=*/false, b,
                /*c_mod=*/(short)0, acc[t], /*reuse_a=*/false, /*reuse_b=*/false);
        }
    }

#pragma unroll
    for (int t = 0; t < NT; ++t) {
        const int col = nBase + t * 16 + lo;
#pragma unroll
        for (int i = 0; i < 8; ++i) {
            const int row = mBase + i + 8 * hi;
            if (row < M) {
                float v = acc[t][i];
                if (Cin) v += Cin[(size_t)row * N + col];
                D[(size_t)row * N + col] = v;
            }
        }
    }
}

// ---------------------------------------------------------------------------
// Scatter-add points into dense voxel grid (sums + counts).
// ---------------------------------------------------------------------------
__global__ void voxelize_kernel(const float* __restrict__ xd,
                                const int* __restrict__ coords,
                                float* __restrict__ sums,
                                float* __restrict__ cnt)
{
    const int tid = blockIdx.x * blockDim.x + threadIdx.x;
    if (tid >= PP * CD) return;
    const int p = tid / CD;
    const int d = tid - p * CD;
    const int a = coords[p * 3 + 0];
    const int b = coords[p * 3 + 1];
    const int c = coords[p * 3 + 2];
    const int flat = (a * GRID + b) * GRID + c;
    atomicAdd(&sums[(size_t)flat * CD + d], xd[tid]);
    if (d == 0) atomicAdd(&cnt[flat], 1.0f);
}

// ---------------------------------------------------------------------------
// Depthwise 3x3x3 conv over voxel means, evaluated only at occupied coords,
// followed by tanh-approximate GELU.  Output layout: pf[p][d].
// ---------------------------------------------------------------------------
__global__ void conv3d_gather_gelu_kernel(const float* __restrict__ sums,
                                          const float* __restrict__ cnt,
                                          const int* __restrict__ coords,
                                          const float* __restrict__ w3,
                                          float* __restrict__ pf)
{
    const int tid = blockIdx.x * blockDim.x + threadIdx.x;
    if (tid >= PP * CD) return;
    const int p = tid / CD;
    const int d = tid - p * CD;
    const int z = coords[p * 3 + 0];
    const int y = coords[p * 3 + 1];
    const int x = coords[p * 3 + 2];

    const float* __restrict__ wk = w3 + d * 27;
    float acc = 0.0f;
#pragma unroll
    for (int kz = 0; kz < 3; ++kz) {
#pragma unroll
        for (int ky = 0; ky < 3; ++ky) {
#pragma unroll
            for (int kx = 0; kx < 3; ++kx) {
                const int zz = z + kz - 1;
                const int yy = y + ky - 1;
                const int xx = x + kx - 1;
                if ((unsigned)zz < (unsigned)GRID &&
                    (unsigned)yy < (unsigned)GRID &&
                    (unsigned)xx < (unsigned)GRID) {
                    const int fl = (zz * GRID + yy) * GRID + xx;
                    const float c = cnt[fl];
                    const float m = sums[(size_t)fl * CD + d] / fmaxf(c, 1.0f);
                    acc += wk[(kz * 3 + ky) * 3 + kx] * m;
                }
            }
        }
    }
    // gelu (tanh approximation)
    const float t = 0.7978845608028654f * (acc + 0.044715f * acc * acc * acc);
    pf[tid] = 0.5f * acc * (1.0f + tanhf(t));
}

// ---------------------------------------------------------------------------
// Depthwise 3x3 2-D conv over the (V,H,W,Cd) point-feature image.
// Input/output layout: [v][h][w][d] (channels innermost -> coalesced).
// ---------------------------------------------------------------------------
__global__ void conv2d_dw_kernel(const float* __restrict__ pf,
                                 const float* __restrict__ w2,
                                 float* __restrict__ sm)
{
    const int tid = blockIdx.x * blockDim.x + threadIdx.x;
    if (tid >= PP * CD) return;
    const int p = tid / CD;
    const int d = tid - p * CD;
    const int v  = p / (HH * WW);
    const int hw = p - v * (HH * WW);
    const int h  = hw / WW;
    const int w  = hw - h * WW;

    const float* __restrict__ wk = w2 + d * 9;
    float acc = 0.0f;
#pragma unroll
    for (int dy = 0; dy < 3; ++dy) {
#pragma unroll
        for (int dx = 0; dx < 3; ++dx) {
            const int hh2 = h + dy - 1;
            const int ww2 = w + dx - 1;
            if ((unsigned)hh2 < (unsigned)HH && (unsigned)ww2 < (unsigned)WW) {
                acc += wk[dy * 3 + dx] *
                       pf[(size_t)((v * HH + hh2) * WW + ww2) * CD + d];
            }
        }
    }
    sm[tid] = acc;
}

// ---------------------------------------------------------------------------
// Launch
// ---------------------------------------------------------------------------
extern "C" void kernel_launch(void* const* d_in, const int* in_sizes, int n_in,
                              void* d_out, int out_size, void* d_ws, size_t ws_size,
                              hipStream_t stream)
{
    const float* hidden  = (const float*)d_in[0];   // (8, 729, 1152)
    const int*   coords  = (const int*)  d_in[1];   // (8, 27, 27, 3)
    const float* w_down  = (const float*)d_in[2];   // (1152, 288)
    const float* conv3dw = (const float*)d_in[3];   // (288, 1, 3, 3, 3)
    const float* conv2dw = (const float*)d_in[4];   // (288, 1, 3, 3)
    const float* w_up    = (const float*)d_in[5];   // (288, 1152)
    float*       out     = (float*)d_out;           // (8, 729, 1152)

    // Workspace partition (floats)
    float* ws   = (float*)d_ws;
    float* xd   = ws;                               // P * CD
    float* sums = xd + (size_t)PP * CD;             // GG3 * CD
    float* cnt  = sums + (size_t)GG3 * CD;          // GG3
    float* pf   = cnt + (size_t)GG3;                // P * CD
    float* sm   = pf + (size_t)PP * CD;             // P * CD

    // Zero the voxel accumulators (capture-safe memset nodes)
    (void)hipMemsetAsync(sums, 0, (size_t)GG3 * CD * sizeof(float), stream);
    (void)hipMemsetAsync(cnt,  0, (size_t)GG3 * sizeof(float), stream);

    constexpr int NT = 6;                 // 16x96 strip per wave
    constexpr int WAVES_PER_BLOCK = 8;    // 256 threads

    // 1) xd = hidden @ w_down   (5832 x 1152) * (1152 x 288)
    {
        const int mTiles  = (PP + 15) / 16;           // 365
        const int nGroups = CD / (16 * NT);           // 3
        const int tiles   = mTiles * nGroups;         // 1095 waves
        const int blocks  = (tiles + WAVES_PER_BLOCK - 1) / WAVES_PER_BLOCK;
        gemm_wmma_f32<NT><<<blocks, 32 * WAVES_PER_BLOCK, 0, stream>>>(
            hidden, w_down, nullptr, xd, PP, CD, CC);
    }

    // 2) voxel scatter-mean accumulation
    {
        const int n = PP * CD;
        voxelize_kernel<<<(n + 255) / 256, 256, 0, stream>>>(xd, coords, sums, cnt);
    }

    // 3) depthwise 3D conv + GELU, gathered at coords
    {
        const int n = PP * CD;
        conv3d_gather_gelu_kernel<<<(n + 255) / 256, 256, 0, stream>>>(
            sums, cnt, coords, conv3dw, pf);
    }

    // 4) depthwise 2D conv
    {
        const int n = PP * CD;
        conv2d_dw_kernel<<<(n + 255) / 256, 256, 0, stream>>>(pf, conv2dw, sm);
    }

    // 5) out = hidden + sm @ w_up   (5832 x 288) * (288 x 1152)
    {
        const int mTiles  = (PP + 15) / 16;           // 365
        const int nGroups = CC / (16 * NT);           // 12
        const int tiles   = mTiles * nGroups;         // 4380 waves
        const int blocks  = (tiles + WAVES_PER_BLOCK - 1) / WAVES_PER_BLOCK;
        gemm_wmma_f32<NT><<<blocks, 32 * WAVES_PER_BLOCK, 0, stream>>>(
            sm, w_up, hidden, out, PP, CC, CD);
    }
}